// PointNet2Encoder_77352361001084
// MI455X (gfx1250) — compile-verified
//
#include <hip/hip_runtime.h>
#include <hip/hip_bf16.h>

typedef float v2f __attribute__((ext_vector_type(2)));
typedef float v8f __attribute__((ext_vector_type(8)));

#define BN_EPS 1e-5f

// ---------------------------------------------------------------------------
// Farthest point sampling: one workgroup per batch. dist[] lives in LDS
// (N <= 4096 -> 16KB of the 320KB WGP LDS). G sequential rounds of
// (update min-dist, block argmax with first-index tie-break) matching
// jnp.argmax semantics.
// ---------------------------------------------------------------------------
__global__ void fps_kernel(const float* __restrict__ pts,
                           float* __restrict__ centers,
                           int N, int G) {
    __shared__ float dist[4096];
    __shared__ float rv[256];
    __shared__ int   ri[256];
    __shared__ float cen[3];
    __shared__ int   far_s;

    const int b   = blockIdx.x;
    const int tid = threadIdx.x;
    const float* P = pts + (size_t)b * N * 3;

    for (int n = tid; n < N; n += 256) dist[n] = 1e10f;
    if (tid == 0) far_s = 0;
    __syncthreads();

    for (int g = 0; g < G; ++g) {
        const int far = far_s;
        if (tid < 3) {
            float c = P[far * 3 + tid];
            centers[((size_t)b * G + g) * 3 + tid] = c;
            cen[tid] = c;
        }
        __syncthreads();
        const float cx = cen[0], cy = cen[1], cz = cen[2];

        float bv = -1.0f; int bi = 0x7fffffff;
        for (int n = tid; n < N; n += 256) {
            float dx = P[n * 3 + 0] - cx;
            float dy = P[n * 3 + 1] - cy;
            float dz = P[n * 3 + 2] - cz;
            float d  = dx * dx + dy * dy + dz * dz;
            float nd = fminf(dist[n], d);
            dist[n] = nd;
            if (nd > bv) { bv = nd; bi = n; }   // strict > keeps lowest index
        }
        rv[tid] = bv; ri[tid] = bi;
        __syncthreads();
        for (int s = 128; s > 0; s >>= 1) {
            if (tid < s) {
                float ov = rv[tid + s]; int oi = ri[tid + s];
                if (ov > rv[tid] || (ov == rv[tid] && oi < ri[tid])) {
                    rv[tid] = ov; ri[tid] = oi;
                }
            }
            __syncthreads();
        }
        if (tid == 0) far_s = ri[0];
        __syncthreads();
    }
}

// ---------------------------------------------------------------------------
// Ball query: one thread per (b,g). Scan points in index order, keep first K
// within radius, pad with first hit (center itself is always a hit).
// ---------------------------------------------------------------------------
__global__ void ball_query_kernel(const float* __restrict__ pts,
                                  const float* __restrict__ centers,
                                  int* __restrict__ idx,
                                  int B, int N, int G, int K, float r2) {
    const int t = blockIdx.x * blockDim.x + threadIdx.x;
    if (t >= B * G) return;
    const int b = t / G;
    const float* P = pts + (size_t)b * N * 3;
    const float cx = centers[t * 3 + 0];
    const float cy = centers[t * 3 + 1];
    const float cz = centers[t * 3 + 2];
    int* out = idx + (size_t)t * K;

    int cnt = 0;
    for (int n = 0; n < N && cnt < K; ++n) {
        float dx = P[n * 3 + 0] - cx;
        float dy = P[n * 3 + 1] - cy;
        float dz = P[n * 3 + 2] - cz;
        if (dx * dx + dy * dy + dz * dz <= r2) out[cnt++] = n;
    }
    const int first = (cnt > 0) ? out[0] : N;
    for (int j = cnt; j < K; ++j) out[j] = first;
}

// ---------------------------------------------------------------------------
// Weight padding: copy (Cout, Cin) row-major weights into (Cout, ldw) with
// zero fill so the GEMM K-loop needs no bounds checks.
// ---------------------------------------------------------------------------
__global__ void pad_weight_kernel(const float* __restrict__ W,
                                  float* __restrict__ Wp,
                                  int Cout, int Cin, int ldw) {
    const int t = blockIdx.x * blockDim.x + threadIdx.x;
    if (t >= Cout * ldw) return;
    const int o = t / ldw;
    const int c = t % ldw;
    Wp[t] = (c < Cin) ? W[(size_t)o * Cin + c] : 0.f;
}

// ---------------------------------------------------------------------------
// Gather stages (write with padded leading dimension, zero the pad columns)
// ---------------------------------------------------------------------------
__global__ void gather_sa1_kernel(const float* __restrict__ pts,
                                  const int* __restrict__ idx,
                                  const float* __restrict__ centers,
                                  float* __restrict__ X,      // ldx = 4
                                  int B, int N, int G, int K) {
    const int t = blockIdx.x * blockDim.x + threadIdx.x;
    if (t >= B * G * K) return;
    const int bg = t / K;
    const int b  = bg / G;
    const int id = idx[t];
    const float* p = pts + ((size_t)b * N + id) * 3;
    const float* c = centers + (size_t)bg * 3;
    float* xo = X + (size_t)t * 4;
    xo[0] = p[0] - c[0];
    xo[1] = p[1] - c[1];
    xo[2] = p[2] - c[2];
    xo[3] = 0.f;                                  // K-pad
}

__global__ void gather_sa2_kernel(const float* __restrict__ ppts,   // (B,N1,3)
                                  const float* __restrict__ feats,  // (B,N1,C)
                                  const int* __restrict__ idx,
                                  const float* __restrict__ centers,
                                  float* __restrict__ X,            // ldx = 3+C+1
                                  int B, int N1, int G, int K, int C) {
    const int t = blockIdx.x * blockDim.x + threadIdx.x;
    if (t >= B * G * K) return;
    const int bg = t / K;
    const int b  = bg / G;
    const int id = idx[t];
    const int ldx = 3 + C + 1;
    const float* p = ppts + ((size_t)b * N1 + id) * 3;
    const float* c = centers + (size_t)bg * 3;
    float* xo = X + (size_t)t * ldx;
    xo[0] = p[0] - c[0];
    xo[1] = p[1] - c[1];
    xo[2] = p[2] - c[2];
    const float* f = feats + ((size_t)b * N1 + id) * C;
    for (int j = 0; j < C; ++j) xo[3 + j] = f[j];
    xo[3 + C] = 0.f;                              // K-pad
}

__global__ void gather_sa3_kernel(const float* __restrict__ ppts,  // (B,Np,3)
                                  const float* __restrict__ feats, // (B,Np,C)
                                  float* __restrict__ X,           // ldx = 3+C+1
                                  int B, int Np, int C) {
    const int t = blockIdx.x * blockDim.x + threadIdx.x;
    if (t >= B * Np) return;
    const int ldx = 3 + C + 1;
    float* xo = X + (size_t)t * ldx;
    xo[0] = ppts[(size_t)t * 3 + 0];
    xo[1] = ppts[(size_t)t * 3 + 1];
    xo[2] = ppts[(size_t)t * 3 + 2];
    const float* f = feats + (size_t)t * C;
    for (int j = 0; j < C; ++j) xo[3 + j] = f[j];
    xo[3 + C] = 0.f;                              // K-pad
}

// ---------------------------------------------------------------------------
// Fused GEMM + BatchNorm + ReLU using V_WMMA_F32_16X16X4_F32.
//   Y[m][o] = relu((sum_c X[m][c] W[o][c] + b[o] - mu[o])*g[o]*rsqrt(var+eps) + beta[o])
// One wave per 16(M)x16(Cout) tile. ISA 7.12.2 f32 layouts:
//   A 16x4: lane<16 -> M=lane, a[j] = K(2*half+j); half = lane>>4
//   B 4x16: b[j]    = B[K=2*half+j][N=lane&15] = W[col][k]
//   C/D:    acc[r] -> M = r + 8*half, N = lane&15
// Kpad is a multiple of 4 and both X (ldx) and W (ldw) are zero-padded, so
// the inner loop is two unconditional 8-byte loads + one WMMA (no EXEC
// masking; EXEC stays all-ones as WMMA requires — the wid guard is
// wave-uniform). Alignment: ldx/ldw, 2*half and kk are all even -> 8B.
// ---------------------------------------------------------------------------
__global__ void wmma_gemm_bn_relu_kernel(const float* __restrict__ X, int ldx,
                                         const float* __restrict__ W, int ldw,
                                         const float* __restrict__ bias,
                                         const float* __restrict__ gamma,
                                         const float* __restrict__ beta,
                                         const float* __restrict__ mean,
                                         const float* __restrict__ var,
                                         float* __restrict__ Y,
                                         int M, int Kpad, int Cout) {
    const int lane = threadIdx.x & 31;
    const int wid  = (blockIdx.x * blockDim.x + threadIdx.x) >> 5;
    const int tiles_m = M >> 4;
    const int tiles_n = Cout >> 4;
    if (wid >= tiles_m * tiles_n) return;

    const int tm = wid % tiles_m;
    const int tn = wid / tiles_m;
    const int row0 = tm << 4;
    const int col0 = tn << 4;
    const int half = lane >> 4;
    const int lm   = lane & 15;

    const float* xp = X + (size_t)(row0 + lm) * ldx + 2 * half;
    const float* wp = W + (size_t)(col0 + lm) * ldw + 2 * half;

    v8f acc = {0.f, 0.f, 0.f, 0.f, 0.f, 0.f, 0.f, 0.f};
    for (int kk = 0; kk < Kpad; kk += 4) {
        const v2f a  = *(const v2f*)(xp + kk);
        const v2f bf = *(const v2f*)(wp + kk);
        acc = __builtin_amdgcn_wmma_f32_16x16x4_f32(
            /*neg_a=*/false, a, /*neg_b=*/false, bf,
            /*c_mod=*/(short)0, acc, /*reuse_a=*/false, /*reuse_b=*/false);
    }

    const int ch = col0 + lm;
    const float s  = gamma[ch] * rsqrtf(var[ch] + BN_EPS);
    const float sh = (bias[ch] - mean[ch]) * s + beta[ch];
#pragma unroll
    for (int r = 0; r < 8; ++r) {
        const int mrow = row0 + r + 8 * half;
        float y = acc[r] * s + sh;
        y = fmaxf(y, 0.f);
        Y[(size_t)mrow * Cout + ch] = y;
    }
}

// ---------------------------------------------------------------------------
// Max over the K (neighbor) axis: Y (BG,K,C) -> F (BG,C)
// ---------------------------------------------------------------------------
__global__ void max_over_k_kernel(const float* __restrict__ Y,
                                  float* __restrict__ F,
                                  int BG, int K, int C) {
    const int t = blockIdx.x * blockDim.x + threadIdx.x;
    if (t >= BG * C) return;
    const int bg = t / C;
    const int c  = t % C;
    const float* y = Y + (size_t)bg * K * C + c;
    float m = y[0];
    for (int k = 1; k < K; ++k) m = fmaxf(m, y[(size_t)k * C]);
    F[t] = m;
}

// ---------------------------------------------------------------------------
// Host orchestration
// ---------------------------------------------------------------------------
extern "C" void kernel_launch(void* const* d_in, const int* in_sizes, int n_in,
                              void* d_out, int out_size, void* d_ws, size_t ws_size,
                              hipStream_t stream) {
    (void)in_sizes; (void)n_in; (void)out_size; (void)ws_size;

    const float* x = (const float*)d_in[0];
    // params p (0..2), layer l (0..2), field f: 0=W 1=b 2=g 3=beta 4=mu 5=var
    auto P = [&](int p, int l, int f) -> const float* {
        return (const float*)d_in[1 + p * 18 + l * 6 + f];
    };

    const int B = 16, N = 4096;
    const int G1 = 512, K1 = 32, C1 = 128;
    const int G2 = 128, K2 = 64, C2 = 256;
    const int K3 = G2;                 // 128 "neighbors" in group-all stage
    const int LD1 = 4, LD2 = 132, LD3 = 260;   // padded K strides

    // ---- carve workspace (256B aligned) ----
    char* wsp = (char*)d_ws;
    size_t off = 0;
    auto alloc = [&](size_t bytes) -> void* {
        void* p = wsp + off;
        off = (off + bytes + 255) & ~(size_t)255;
        return p;
    };
    float* c1   = (float*)alloc((size_t)B * G1 * 3 * sizeof(float));
    int*   idx1 = (int*)  alloc((size_t)B * G1 * K1 * sizeof(int));
    float* X1   = (float*)alloc((size_t)B * G1 * K1 * LD1 * sizeof(float));
    float* f1   = (float*)alloc((size_t)B * G1 * C1 * sizeof(float));
    float* c2   = (float*)alloc((size_t)B * G2 * 3 * sizeof(float));
    int*   idx2 = (int*)  alloc((size_t)B * G2 * K2 * sizeof(int));
    float* f2   = (float*)alloc((size_t)B * G2 * C2 * sizeof(float));
    float* X3   = (float*)alloc((size_t)B * K3 * LD3 * sizeof(float));
    float* Wp0  = (float*)alloc((size_t)64  * LD1 * sizeof(float));   // SA1 L0
    float* Wp1  = (float*)alloc((size_t)128 * LD2 * sizeof(float));   // SA2 L0
    float* Wp2  = (float*)alloc((size_t)256 * LD3 * sizeof(float));   // SA3 L0
    const size_t bufElems = (size_t)B * G1 * K1 * 128;   // 33.5M floats
    float* bufA = (float*)alloc(bufElems * sizeof(float));
    float* bufB = (float*)alloc(bufElems * sizeof(float));

    auto gemm = [&](const float* Xp, int ldx, const float* Wq, int ldw,
                    int p, int l, float* Yp, int M, int Kpad, int Cout) {
        const long long waves = (long long)(M / 16) * (Cout / 16);
        const long long threads = waves * 32;
        const int block = 256;
        const int grid = (int)((threads + block - 1) / block);
        wmma_gemm_bn_relu_kernel<<<grid, block, 0, stream>>>(
            Xp, ldx, Wq, ldw, P(p, l, 1), P(p, l, 2), P(p, l, 3), P(p, l, 4),
            P(p, l, 5), Yp, M, Kpad, Cout);
    };

    // ---- pad the three ragged-K weight matrices once ----
    pad_weight_kernel<<<(64 * LD1 + 255) / 256, 256, 0, stream>>>(
        P(0, 0, 0), Wp0, 64, 3, LD1);
    pad_weight_kernel<<<(128 * LD2 + 255) / 256, 256, 0, stream>>>(
        P(1, 0, 0), Wp1, 128, 131, LD2);
    pad_weight_kernel<<<(256 * LD3 + 255) / 256, 256, 0, stream>>>(
        P(2, 0, 0), Wp2, 256, 259, LD3);

    // ================= SA1 =================
    fps_kernel<<<B, 256, 0, stream>>>(x, c1, N, G1);
    {
        const int ng = B * G1;
        ball_query_kernel<<<(ng + 255) / 256, 256, 0, stream>>>(
            x, c1, idx1, B, N, G1, K1, 0.2f * 0.2f);
    }
    const int M1 = B * G1 * K1;  // 262144
    gather_sa1_kernel<<<(M1 + 255) / 256, 256, 0, stream>>>(
        x, idx1, c1, X1, B, N, G1, K1);
    gemm(X1,   LD1, Wp0,        LD1, 0, 0, bufA, M1, LD1, 64);
    gemm(bufA, 64,  P(0, 1, 0), 64,  0, 1, bufB, M1, 64,  64);
    gemm(bufB, 64,  P(0, 2, 0), 64,  0, 2, bufA, M1, 64,  128);
    {
        const int nt = B * G1 * C1;
        max_over_k_kernel<<<(nt + 255) / 256, 256, 0, stream>>>(
            bufA, f1, B * G1, K1, C1);
    }

    // ================= SA2 =================
    fps_kernel<<<B, 256, 0, stream>>>(c1, c2, G1, G2);
    {
        const int ng = B * G2;
        ball_query_kernel<<<(ng + 255) / 256, 256, 0, stream>>>(
            c1, c2, idx2, B, G1, G2, K2, 0.4f * 0.4f);
    }
    const int M2 = B * G2 * K2;  // 131072
    float* X2 = bufB;            // 131072*132 floats fits in bufB
    gather_sa2_kernel<<<(M2 + 255) / 256, 256, 0, stream>>>(
        c1, f1, idx2, c2, X2, B, G1, G2, K2, C1);
    gemm(X2,   LD2, Wp1,        LD2, 1, 0, bufA, M2, LD2, 128);
    gemm(bufA, 128, P(1, 1, 0), 128, 1, 1, bufB, M2, 128, 128);
    gemm(bufB, 128, P(1, 2, 0), 128, 1, 2, bufA, M2, 128, 256);
    {
        const int nt = B * G2 * C2;
        max_over_k_kernel<<<(nt + 255) / 256, 256, 0, stream>>>(
            bufA, f2, B * G2, K2, C2);
    }

    // ================= SA3 (group-all) =================
    const int M3 = B * K3;  // 2048
    gather_sa3_kernel<<<(M3 + 255) / 256, 256, 0, stream>>>(
        c2, f2, X3, B, K3, C2);
    gemm(X3,   LD3, Wp2,        LD3, 2, 0, bufB, M3, LD3, 256);
    gemm(bufB, 256, P(2, 1, 0), 256, 2, 1, bufA, M3, 256, 512);
    gemm(bufA, 512, P(2, 2, 0), 512, 2, 2, bufB, M3, 512, 1024);
    {
        const int nt = B * 1024;
        max_over_k_kernel<<<(nt + 255) / 256, 256, 0, stream>>>(
            bufB, (float*)d_out, B, K3, 1024);
    }
}